// DNeRFTensoRF_40681930227974
// MI455X (gfx1250) — compile-verified
//
#include <hip/hip_runtime.h>

typedef __attribute__((ext_vector_type(2))) float v2f;
typedef __attribute__((ext_vector_type(8))) float v8f;

#define NFRAMES 100
#define RESO    256
#define CHANK   64
#define FREQK   10
#define PLANE_HW (NFRAMES * RESO)      /* 25600 floats per (channel) plane  */
#define PLANE_SZ (CHANK * PLANE_HW)    /* 1,638,400 floats per spatial axis */
#define INV_2PI  0.15915494309189535f

__global__ __launch_bounds__(256) void dnerf_fused(
    const float* __restrict__ X,
    const float* __restrict__ F0,
    const float* __restrict__ F1,
    const float* __restrict__ F2,
    float* __restrict__ out)
{
  const int pid  = blockIdx.x * 256 + (int)threadIdx.x;
  const int lane = (int)threadIdx.x & 31;

  // ---- point (P,4), 16B aligned -> single b128 load ----
  const float4 xin = *(const float4*)(X + (size_t)pid * 4);
  const float xyz[3] = { xin.x, xin.y, xin.z };
  const float t = xin.w;

  // ---- temporal coordinate (shared across planes & features) ----
  const float tn  = t / (float)(NFRAMES - 1);
  const float gy  = 2.0f * tn - 1.0f;
  const float iy  = (gy + 1.0f) * 0.5f * (float)(NFRAMES - 1);
  const float y0f = floorf(iy);
  const int   y0i = (int)y0f;
  const int   y1i = y0i + 1;
  const float wy1 = iy - y0f;
  const float wy0 = 1.0f - wy1;
  const float iny0 = (y0i >= 0 && y0i < NFRAMES) ? 1.0f : 0.0f;
  const float iny1 = (y1i >= 0 && y1i < NFRAMES) ? 1.0f : 0.0f;
  const int   y0c = min(max(y0i, 0), NFRAMES - 1);
  const int   y1c = min(max(y1i, 0), NFRAMES - 1);

  // ---- per-plane corner offsets + mask-folded weights (hoisted out of c loop) ----
  int   off[3][4];
  float wgt[3][4];
#pragma unroll
  for (int i = 0; i < 3; ++i) {
    const float gx  = 2.0f * xyz[i] - 1.0f;
    const float ix  = (gx + 1.0f) * 0.5f * (float)(RESO - 1);
    const float x0f = floorf(ix);
    const int   x0i = (int)x0f;
    const int   x1i = x0i + 1;
    const float wx1 = ix - x0f;
    const float wx0 = 1.0f - wx1;
    const float inx0 = (x0i >= 0 && x0i < RESO) ? 1.0f : 0.0f;
    const float inx1 = (x1i >= 0 && x1i < RESO) ? 1.0f : 0.0f;
    const int   x0c = min(max(x0i, 0), RESO - 1);
    const int   x1c = min(max(x1i, 0), RESO - 1);
    off[i][0] = y0c * RESO + x0c;  wgt[i][0] = wy0 * wx0 * iny0 * inx0;
    off[i][1] = y0c * RESO + x1c;  wgt[i][1] = wy0 * wx1 * iny0 * inx1;
    off[i][2] = y1c * RESO + x0c;  wgt[i][2] = wy1 * wx0 * iny1 * inx0;
    off[i][3] = y1c * RESO + x1c;  wgt[i][3] = wy1 * wx1 * iny1 * inx1;
  }

  // ---- channel loop: 36 independent L2 gathers + triple product per channel ----
  const float* __restrict__ bases[3] = { F0, F1, F2 };
  float acc[3] = { 0.0f, 0.0f, 0.0f };
#pragma unroll 2
  for (int c = 0; c < CHANK; ++c) {
    const int cb = c * PLANE_HW;
#pragma unroll
    for (int f = 0; f < 3; ++f) {
      float s[3];
#pragma unroll
      for (int i = 0; i < 3; ++i) {
        const float* __restrict__ bp = bases[f] + (size_t)i * PLANE_SZ + cb;
        s[i] = wgt[i][0] * bp[off[i][0]] + wgt[i][1] * bp[off[i][1]]
             + wgt[i][2] * bp[off[i][2]] + wgt[i][3] * bp[off[i][3]];
      }
      acc[f] += s[0] * s[1] * s[2];
    }
  }

  const float p0 = xyz[0] + acc[0];
  const float p1 = xyz[1] + acc[1];
  const float p2 = xyz[2] + acc[2];

  {
    float* __restrict__ orow = out + (size_t)pid * 63;
    orow[0] = p0; orow[1] = p1; orow[2] = p2;
  }

  // ---- positional encoding via WMMA ----------------------------------------
  // ang'[p, n] = p_d * (2^f / 2pi), n = f*3+d  -> v_sin/v_cos take ang' raw.
  // A(16x4) = [px,py,pz,0] per point; B(4x16): B[k][n] = (k==n%3) ? 2^(n/3)/2pi : 0
  // B layout (4x16, wave32): VGPR0 = rows K=0 (lanes 0-15) / K=2 (lanes 16-31),
  //                          VGPR1 = rows K=1 / K=3.
  v2f btile[2];
#pragma unroll
  for (int tile = 0; tile < 2; ++tile) {
    const int n  = tile * 16 + (lane & 15);
    const int k0 = (lane < 16) ? 0 : 2;
    const int dd = n % 3;
    const int ff = n / 3;
    const float fv = (n < 3 * FREQK) ? (float)(1 << ff) * INV_2PI : 0.0f;
    v2f b;
    b.x = (k0     == dd) ? fv : 0.0f;
    b.y = (k0 + 1 == dd) ? fv : 0.0f;
    btile[tile] = b;
  }

  const int wave_pt0 = pid - lane;  // first point of this wave
#pragma unroll
  for (int h = 0; h < 2; ++h) {     // 16-point halves of the wave
    // A layout (16x4): lanes 0-15 carry K=0,1 (px,py); lanes 16-31 carry K=2,3 (pz,0)
    const int src = h * 16 + (lane & 15);
    const float ax = __shfl(p0, src, 32);
    const float ay = __shfl(p1, src, 32);
    const float az = __shfl(p2, src, 32);
    v2f a;
    a.x = (lane < 16) ? ax : az;
    a.y = (lane < 16) ? ay : 0.0f;

#pragma unroll
    for (int tile = 0; tile < 2; ++tile) {
      v8f dmat = { 0.f, 0.f, 0.f, 0.f, 0.f, 0.f, 0.f, 0.f };
      dmat = __builtin_amdgcn_wmma_f32_16x16x4_f32(
          /*neg_a=*/false, a, /*neg_b=*/false, btile[tile],
          /*c_mod=*/(short)0, dmat, /*reuse_a=*/false, /*reuse_b=*/false);

      // D layout: lane l holds rows (r + 8*(l>=16)), col l%16 in VGPR r.
      const int n = tile * 16 + (lane & 15);
      if (n < 3 * FREQK) {
        const int dd = n % 3;
        const int ff = n / 3;
        const int cs = 3 + ff * 6 + dd;       // sin column; cos = cs + 3
        const int rbase = (lane < 16) ? 0 : 8;
        // Single base address per (h,tile); rows are a constant 63-float stride
        // -> all 16 stores use instruction immediate offsets off one VGPR pair.
        float* __restrict__ rowbase =
            out + (size_t)(wave_pt0 + h * 16 + rbase) * 63 + cs;
#pragma unroll
        for (int r = 0; r < 8; ++r) {
          const float ang = dmat[r];
          rowbase[r * 63]     = __builtin_amdgcn_sinf(ang);
          rowbase[r * 63 + 3] = __builtin_amdgcn_cosf(ang);
        }
      }
    }
  }
}

extern "C" void kernel_launch(void* const* d_in, const int* in_sizes, int n_in,
                              void* d_out, int out_size, void* d_ws, size_t ws_size,
                              hipStream_t stream) {
  (void)n_in; (void)out_size; (void)d_ws; (void)ws_size;
  const float* X  = (const float*)d_in[0];
  const float* F0 = (const float*)d_in[1];
  const float* F1 = (const float*)d_in[2];
  const float* F2 = (const float*)d_in[3];
  float* out = (float*)d_out;
  const int P = in_sizes[0] / 4;          // (P,4) points; P = 524288
  const int blocks = P / 256;             // exact: 2048 blocks of 8 waves
  dnerf_fused<<<blocks, 256, 0, stream>>>(X, F0, F1, F2, out);
}